// GCN_Sparse_Policy_5_30528627540627
// MI455X (gfx1250) — compile-verified
//
#include <hip/hip_runtime.h>
#include <hip/hip_bf16.h>
#include <math.h>

// ---------------------------------------------------------------------------
// GCN (5 layers) for MI455X / gfx1250, wave32.
//   h = relu(spmm(A, h@W) + b)  x4 ;  h5 = spmm(A, h@W5) + b5
//   out = log_softmax(h5, axis=0)   (over nodes, per class)
// GEMMs use V_WMMA_F32_16X16X4_F32 (fp32 matrix path -> matches fp32 ref).
// SpMM is node-parallel, coalesced, L2-resident gather (+ global prefetch).
// ---------------------------------------------------------------------------

typedef float v2f __attribute__((ext_vector_type(2)));
typedef float v8f __attribute__((ext_vector_type(8)));

#define NNODES 50000
#define NEDGES 1600000
#define KDIM   256          // NIN == NHID == 256 (K of every GEMM)
#define M_BLK  80           // 5 WMMA M-tiles per wave; 50000 % 80 == 0
#define LDS_PITCH 260       // 256 + 4 -> bank shift of 4 per row (conflict-free b64)

// ---------------------------------------------------------------------------
// Build CSR row_ptr from sorted COO rows: rp[n] = first edge with row >= n.
// ---------------------------------------------------------------------------
__global__ void build_rowptr(const int* __restrict__ row, int* __restrict__ rp,
                             int E, int N) {
  int i = blockIdx.x * blockDim.x + threadIdx.x;
  if (i >= E) return;
  if (i == 0) {
    for (int r = 0; r <= row[0]; ++r) rp[r] = 0;
  } else if (row[i] != row[i - 1]) {
    for (int r = row[i - 1] + 1; r <= row[i]; ++r) rp[r] = i;
  }
  if (i == E - 1) {
    for (int r = row[E - 1] + 1; r <= N; ++r) rp[r] = E;
  }
}

// ---------------------------------------------------------------------------
// C[M x N] = A[M x 256] * B[256 x N], all fp32 row-major, N multiple of 16.
// Block: (N/16) waves. Each block: 80 rows of A staged in LDS; each wave owns
// a 16-col slice of C and produces 5 stacked 16x16 tiles.
// WMMA fragment layouts per CDNA5 ISA 7.12.2:
//   A 16x4 : lane L, m = L&15, holds K = {kh, kh+1}, kh = (L>>4)*2
//   B 4x16 : lane L, n = L&15, holds rows K = {kh, kh+1}
//   C 16x16: VGPR v, lane L -> (m = v + (L>>4)*8, n = L&15)
// ---------------------------------------------------------------------------
__global__ __launch_bounds__(512) void gemm_wmma_f32(
    const float* __restrict__ A, const float* __restrict__ B,
    float* __restrict__ C, int N) {
  __shared__ float As[M_BLK * LDS_PITCH];

  const int m0 = blockIdx.x * M_BLK;

  // Cooperative, coalesced load of the 80x256 A tile into LDS.
  for (int idx = threadIdx.x; idx < M_BLK * KDIM; idx += blockDim.x) {
    int r = idx >> 8;      // / 256
    int c = idx & 255;
    As[r * LDS_PITCH + c] = A[(size_t)(m0 + r) * KDIM + c];
  }
  __syncthreads();

  const int lane  = threadIdx.x & 31;
  const int wave  = threadIdx.x >> 5;
  const int n0    = wave * 16;
  const int mrow  = lane & 15;
  const int ncol  = lane & 15;
  const int khalf = (lane >> 4) << 1;   // 0 or 2

  v8f acc[5];
#pragma unroll
  for (int t = 0; t < 5; ++t)
#pragma unroll
    for (int i = 0; i < 8; ++i) acc[t][i] = 0.0f;

#pragma unroll 4
  for (int k = 0; k < KDIM; k += 4) {
    v2f b;
    const float* bp = B + (size_t)(k + khalf) * N + n0 + ncol;
    b.x = bp[0];
    b.y = bp[N];
#pragma unroll
    for (int t = 0; t < 5; ++t) {
      v2f a;
      const float* ap = &As[(t * 16 + mrow) * LDS_PITCH + k + khalf];
      a.x = ap[0];
      a.y = ap[1];
      acc[t] = __builtin_amdgcn_wmma_f32_16x16x4_f32(
          /*neg_a=*/false, a, /*neg_b=*/false, b,
          /*c_mod=*/(short)0, acc[t], /*reuse_a=*/false, /*reuse_b=*/false);
    }
  }

  const int mhalf = (lane >> 4) << 3;   // 0 or 8
#pragma unroll
  for (int t = 0; t < 5; ++t)
#pragma unroll
    for (int v = 0; v < 8; ++v)
      C[(size_t)(m0 + t * 16 + mhalf + v) * N + n0 + ncol] = acc[t][v];
}

// ---------------------------------------------------------------------------
// out[n, :] = act( sum_e val[e] * sup[col[e], :] + bias )  — one block per
// node, thread t owns column t (fully coalesced 1KB row reads; L2-resident).
// ---------------------------------------------------------------------------
__global__ __launch_bounds__(256) void spmm_bias_act(
    const float* __restrict__ sup, const int* __restrict__ rp,
    const int* __restrict__ colIdx, const float* __restrict__ val,
    const float* __restrict__ bias, float* __restrict__ out,
    int ld, int do_relu) {
  const int n = blockIdx.x;
  const int t = threadIdx.x;
  const int e0 = rp[n], e1 = rp[n + 1];
  float acc = 0.0f;
  for (int e = e0; e < e1; ++e) {
    const int c = colIdx[e];
    const float v = val[e];
    if (e + 1 < e1) {
      // hide L2 latency of the next gathered row
      __builtin_prefetch(&sup[(size_t)colIdx[e + 1] * ld + t], 0, 3);
    }
    acc = fmaf(sup[(size_t)c * ld + t], v, acc);
  }
  acc += bias[t];
  if (do_relu) acc = fmaxf(acc, 0.0f);
  out[(size_t)n * ld + t] = acc;
}

// ---------------------------------------------------------------------------
// Zero-pad W5 [256 x 2] -> [256 x 16] so the WMMA GEMM can produce support5.
// ---------------------------------------------------------------------------
__global__ void pad_w5(const float* __restrict__ W5, float* __restrict__ W5p) {
  int i = blockIdx.x * blockDim.x + threadIdx.x;
  if (i < KDIM * 16) {
    int r = i >> 4, c = i & 15;
    W5p[i] = (c < 2) ? W5[r * 2 + c] : 0.0f;
  }
}

// ---------------------------------------------------------------------------
// Final SpMM on 2 classes: h5[n, c] = sum_e val[e]*sup5[col[e], c] + b5[c].
// sup5 has leading dim 16 (padded GEMM output).
// ---------------------------------------------------------------------------
__global__ void spmm5(const float* __restrict__ sup5, const int* __restrict__ rp,
                      const int* __restrict__ colIdx, const float* __restrict__ val,
                      const float* __restrict__ b5, float* __restrict__ h5) {
  int n = blockIdx.x * blockDim.x + threadIdx.x;
  if (n >= NNODES) return;
  const int e0 = rp[n], e1 = rp[n + 1];
  float a0 = 0.0f, a1 = 0.0f;
  for (int e = e0; e < e1; ++e) {
    const int c = colIdx[e];
    const float v = val[e];
    a0 = fmaf(sup5[(size_t)c * 16 + 0], v, a0);
    a1 = fmaf(sup5[(size_t)c * 16 + 1], v, a1);
  }
  h5[(size_t)n * 2 + 0] = a0 + b5[0];
  h5[(size_t)n * 2 + 1] = a1 + b5[1];
}

// ---------------------------------------------------------------------------
// log_softmax over axis=0 (nodes), per class c:
//   stats[c] = max_n h[n,c] + log(sum_n exp(h[n,c]-max))
// Single 1024-thread block; 50000x2 elements is trivial reduction traffic.
// ---------------------------------------------------------------------------
__global__ __launch_bounds__(1024) void lse_reduce(
    const float* __restrict__ h, float* __restrict__ stats, int N) {
  __shared__ float red[1024];
  const int t = threadIdx.x;
  for (int c = 0; c < 2; ++c) {
    float m = -INFINITY;
    for (int n = t; n < N; n += 1024) m = fmaxf(m, h[(size_t)n * 2 + c]);
    red[t] = m;
    __syncthreads();
    for (int s = 512; s > 0; s >>= 1) {
      if (t < s) red[t] = fmaxf(red[t], red[t + s]);
      __syncthreads();
    }
    const float gmax = red[0];
    __syncthreads();
    float sum = 0.0f;
    for (int n = t; n < N; n += 1024) sum += expf(h[(size_t)n * 2 + c] - gmax);
    red[t] = sum;
    __syncthreads();
    for (int s = 512; s > 0; s >>= 1) {
      if (t < s) red[t] += red[t + s];
      __syncthreads();
    }
    if (t == 0) stats[c] = gmax + logf(red[0]);
    __syncthreads();
  }
}

__global__ void lse_apply(const float* __restrict__ h,
                          const float* __restrict__ stats,
                          float* __restrict__ out, int N) {
  int i = blockIdx.x * blockDim.x + threadIdx.x;
  if (i < N * 2) out[i] = h[i] - stats[i & 1];
}

// ---------------------------------------------------------------------------
extern "C" void kernel_launch(void* const* d_in, const int* in_sizes, int n_in,
                              void* d_out, int out_size, void* d_ws, size_t ws_size,
                              hipStream_t stream) {
  const float* feat = (const float*)d_in[0];
  const int*   row  = (const int*)d_in[1];
  const int*   col  = (const int*)d_in[2];
  const float* val  = (const float*)d_in[3];
  const float* W[5] = {(const float*)d_in[4],  (const float*)d_in[6],
                       (const float*)d_in[8],  (const float*)d_in[10],
                       (const float*)d_in[12]};
  const float* b[5] = {(const float*)d_in[5],  (const float*)d_in[7],
                       (const float*)d_in[9],  (const float*)d_in[11],
                       (const float*)d_in[13]};

  char* wsb = (char*)d_ws;
  size_t off = 0;
  auto take = [&](size_t bytes) -> void* {
    void* p = wsb + off;
    off += (bytes + 255) & ~(size_t)255;
    return p;
  };
  int*   rp    = (int*)  take((NNODES + 1) * sizeof(int));
  float* h     = (float*)take((size_t)NNODES * KDIM * sizeof(float));
  float* s     = (float*)take((size_t)NNODES * KDIM * sizeof(float));
  float* w5p   = (float*)take((size_t)KDIM * 16 * sizeof(float));
  float* sup5  = (float*)take((size_t)NNODES * 16 * sizeof(float));
  float* h5    = (float*)take((size_t)NNODES * 2 * sizeof(float));
  float* stats = (float*)take(2 * sizeof(float));
  (void)ws_size; (void)in_sizes; (void)n_in; (void)out_size;

  build_rowptr<<<(NEDGES + 255) / 256, 256, 0, stream>>>(row, rp, NEDGES, NNODES);

  const float* cur = feat;
  for (int l = 0; l < 4; ++l) {
    gemm_wmma_f32<<<NNODES / M_BLK, 512, 0, stream>>>(cur, W[l], s, KDIM);
    spmm_bias_act<<<NNODES, 256, 0, stream>>>(s, rp, col, val, b[l], h, KDIM, 1);
    cur = h;
  }

  pad_w5<<<(KDIM * 16 + 255) / 256, 256, 0, stream>>>(W[4], w5p);
  gemm_wmma_f32<<<NNODES / M_BLK, 32, 0, stream>>>(h, w5p, sup5, 16);
  spmm5<<<(NNODES + 255) / 256, 256, 0, stream>>>(sup5, rp, col, val, b[4], h5);

  lse_reduce<<<1, 1024, 0, stream>>>(h5, stats, NNODES);
  lse_apply<<<(NNODES * 2 + 255) / 256, 256, 0, stream>>>(h5, stats,
                                                          (float*)d_out, NNODES);
}